// AnchorTargetLayer_49864570306631
// MI455X (gfx1250) — compile-verified
//
#include <hip/hip_runtime.h>
#include <cstdint>

// ---------------------------------------------------------------------------
// AnchorTargetLayer for MI455X (gfx1250, wave32).
// VALU-bound elementwise min/max/mul workload -> no matmul structure, so no
// WMMA; instead we use the CDNA5 async-to-LDS engine (ASYNCcnt) to stage the
// anchor tiles and gt boxes, v_rcp_f32 for the IoU division, and
// float-as-int global atomic max for the cross-block per-gt reduction.
// ---------------------------------------------------------------------------

#define A_N 250000
#define B_N 8
#define G_N 64
#define BG  (B_N * G_N)          // 512 (b,g) pairs

#define POS_T 0.7f
#define NEG_T 0.3f

#define P1_THREADS 256
#define P1_TILE    1024          // anchors staged per block in pass 1
#define P2_THREADS 256

// ---- CDNA5 async global->LDS copy (tracked by ASYNCcnt) -------------------
__device__ __forceinline__ void async_copy_b128(uint32_t lds_off, const void* gaddr) {
    // per-lane: copies 16 bytes from the global address gaddr to the LDS offset lds_off
    asm volatile("global_load_async_to_lds_b128 %0, %1, off"
                 :: "v"(lds_off), "v"(gaddr)
                 : "memory");
}
__device__ __forceinline__ void async_wait_all() {
    asm volatile("s_wait_asynccnt 0" ::: "memory");
}
// Low 32 bits of a flat shared-aperture pointer are the LDS byte offset
// (ISA 10.2).
__device__ __forceinline__ uint32_t lds_off(const void* p) {
    return (uint32_t)(uintptr_t)p;
}

// ---- IoU (one inlined definition so both passes produce identical bits) ---
__device__ __forceinline__ float iou_f(float4 a, float4 g, float area_a, float area_g) {
    float ix1 = fmaxf(a.x, g.x);
    float iy1 = fmaxf(a.y, g.y);
    float ix2 = fminf(a.z, g.z);
    float iy2 = fminf(a.w, g.w);
    float w = fmaxf(ix2 - ix1, 0.0f);
    float h = fmaxf(iy2 - iy1, 0.0f);
    float inter = w * h;
    float denom = area_a + area_g - inter;
    return inter * __builtin_amdgcn_rcpf(denom);   // v_rcp_f32, no div expansion
}

// ---- kernel 0: zero the [B,G] per-gt max table (ws is poisoned 0xAA) ------
__global__ void atl_init_kernel(int* __restrict__ gtmax_i) {
    gtmax_i[threadIdx.x] = 0;                      // 0 bits == +0.0f, IoU >= 0
}

// ---- kernel 1: per-gt max IoU over all anchors ----------------------------
// Block stages a tile of anchors + all gt boxes into LDS via async copies.
// Each thread owns 2 (b,g) pairs in registers and sweeps the tile; anchors
// are read via conflict-free LDS broadcast. One global atomicMax per pair.
__global__ __launch_bounds__(P1_THREADS)
void atl_gtmax_kernel(const float4* __restrict__ anchors,
                      const float4* __restrict__ gt,
                      int* __restrict__ gtmax_i) {
    __shared__ float4 s_anch[P1_TILE];   // 16 KB
    __shared__ float4 s_gt[BG];          //  8 KB

    const int tid = threadIdx.x;
    const int a0  = blockIdx.x * P1_TILE;
    const int n   = min(P1_TILE, A_N - a0);

    for (int i = tid; i < BG; i += P1_THREADS)
        async_copy_b128(lds_off(&s_gt[i]), &gt[i]);
    for (int i = tid; i < n; i += P1_THREADS)
        async_copy_b128(lds_off(&s_anch[i]), &anchors[a0 + i]);
    async_wait_all();
    __syncthreads();

    const float4 g0 = s_gt[tid];
    const float4 g1 = s_gt[tid + P1_THREADS];
    const float ag0 = (g0.z - g0.x) * (g0.w - g0.y);
    const float ag1 = (g1.z - g1.x) * (g1.w - g1.y);

    float m0 = 0.0f, m1 = 0.0f;
    for (int i = 0; i < n; ++i) {
        float4 a = s_anch[i];                       // LDS broadcast
        float aa = (a.z - a.x) * (a.w - a.y);
        m0 = fmaxf(m0, iou_f(a, g0, aa, ag0));
        m1 = fmaxf(m1, iou_f(a, g1, aa, ag1));
    }
    // non-negative floats order correctly as signed ints
    atomicMax(&gtmax_i[tid],              __float_as_int(m0));
    atomicMax(&gtmax_i[tid + P1_THREADS], __float_as_int(m1));
}

// ---- kernel 2: labels + matched boxes -------------------------------------
__global__ __launch_bounds__(P2_THREADS)
void atl_assign_kernel(const float4* __restrict__ anchors,
                       const float4* __restrict__ gt,
                       const int* __restrict__ gtmax_i,
                       float* __restrict__ out_labels,     // [B,A] as float
                       float4* __restrict__ out_matched) { // [B,A,4]
    __shared__ float4 s_gt[BG];      // 8 KB
    __shared__ float  s_gtmax[BG];   // 2 KB

    const int tid = threadIdx.x;
    for (int i = tid; i < BG; i += P2_THREADS)
        async_copy_b128(lds_off(&s_gt[i]), &gt[i]);
    for (int i = tid; i < BG; i += P2_THREADS)
        s_gtmax[i] = __int_as_float(gtmax_i[i]);
    async_wait_all();
    __syncthreads();

    const int a = blockIdx.x * P2_THREADS + tid;
    if (a >= A_N) return;

    const float4 anc = anchors[a];
    const float  aa  = (anc.z - anc.x) * (anc.w - anc.y);

    for (int b = 0; b < B_N; ++b) {
        const float4* gb = &s_gt[b * G_N];
        const float*  mb = &s_gtmax[b * G_N];
        float best = -1.0f;
        int   bestg = 0;
        bool  tie = false;
        #pragma unroll 4
        for (int g = 0; g < G_N; ++g) {
            float4 gg = gb[g];
            float ag = (gg.z - gg.x) * (gg.w - gg.y);
            float v  = iou_f(anc, gg, aa, ag);
            if (v > best) { best = v; bestg = g; }   // first-occurrence argmax
            tie = tie || (v == mb[g]);               // bit-exact recompute
        }
        // sequential where() semantics: neg overrides pos/tie
        float label = (best < NEG_T) ? 0.0f
                    : ((best >= POS_T || tie) ? 1.0f : -1.0f);
        int midx = (best < POS_T) ? 0 : bestg;

        out_labels[(size_t)b * A_N + a]  = label;
        out_matched[(size_t)b * A_N + a] = gb[midx];
    }
}

// ---------------------------------------------------------------------------
extern "C" void kernel_launch(void* const* d_in, const int* in_sizes, int n_in,
                              void* d_out, int out_size, void* d_ws, size_t ws_size,
                              hipStream_t stream) {
    (void)in_sizes; (void)n_in; (void)out_size; (void)ws_size;

    const float4* anchors = (const float4*)d_in[0];   // [A,4] f32
    const float4* gt      = (const float4*)d_in[1];   // [B,G,4] f32

    int*    gtmax_i     = (int*)d_ws;                               // 512 * 4B
    float*  out_labels  = (float*)d_out;                            // [B,A]
    float4* out_matched = (float4*)((float*)d_out + (size_t)B_N * A_N);

    atl_init_kernel<<<1, BG, 0, stream>>>(gtmax_i);

    const int g1 = (A_N + P1_TILE - 1) / P1_TILE;       // 245 blocks
    atl_gtmax_kernel<<<g1, P1_THREADS, 0, stream>>>(anchors, gt, gtmax_i);

    const int g2 = (A_N + P2_THREADS - 1) / P2_THREADS; // 977 blocks
    atl_assign_kernel<<<g2, P2_THREADS, 0, stream>>>(anchors, gt, gtmax_i,
                                                     out_labels, out_matched);
}